// DLP_model2_90555090469432
// MI455X (gfx1250) — compile-verified
//
#include <hip/hip_runtime.h>

typedef __attribute__((ext_vector_type(2))) float v2f;
typedef __attribute__((ext_vector_type(8))) float v8f;

#define EMB_D 128
#define HID   32

// LDS budget per block (256 thr = 8 waves):
//   w1f: 32 ksteps * 2 halves * 32 lanes * 2 floats = 4096 f = 16 KB
//   w2f:  8 ksteps * 2 halves * 32 lanes * 2 floats = 1024 f =  4 KB
//   h1 scratch: 8 waves * 16 rows * 34 (padded)      = 4352 f = 17 KB
__global__ __launch_bounds__(256) void edge_mlp_wmma_f32(
    const int* __restrict__ eidx,   // [2, E]
    const float* __restrict__ emb,  // [N, 128]
    const float* __restrict__ W1, const float* __restrict__ b1,
    const float* __restrict__ W2, const float* __restrict__ b2,
    const float* __restrict__ W3, const float* __restrict__ b3,
    float* __restrict__ out, int E)
{
    __shared__ float w1f[4096];
    __shared__ float w2f[1024];
    __shared__ float h1s[8 * 16 * 34];

    const int tid = threadIdx.x;

    // Pre-swizzle W1/W2 into WMMA B-fragment layout:
    // frag index i = ((s*2 + h)*32 + L)*2 + j  holds  W[k][n],
    //   k = 4s + 2*(L>>4) + j ,  n = 16h + (L&15)
    for (int i = tid; i < 4096; i += 256) {
        int j = i & 1, L = (i >> 1) & 31, h = (i >> 6) & 1, s = i >> 7;
        int k = 4 * s + ((L >> 4) << 1) + j;
        int n = 16 * h + (L & 15);
        w1f[i] = W1[k * HID + n];
    }
    for (int i = tid; i < 1024; i += 256) {
        int j = i & 1, L = (i >> 1) & 31, h = (i >> 6) & 1, s = i >> 7;
        int k = 4 * s + ((L >> 4) << 1) + j;
        int n = 16 * h + (L & 15);
        w2f[i] = W2[k * HID + n];
    }
    __syncthreads();

    const int lane = tid & 31;
    const int wave = tid >> 5;
    const int m = lane & 15;      // edge row within tile / N column within half
    const int g = lane >> 4;      // lane-group: K sub-offset for A/B frags
    float* h1 = &h1s[wave * 16 * 34];

    // per-lane bias / final-layer weights (N = 16h + (lane&15))
    const float b1lo = b1[m],  b1hi = b1[16 + m];
    const float b2lo = b2[m],  b2hi = b2[16 + m];
    const float w3lo = W3[m],  w3hi = W3[16 + m];
    const float b3s  = b3[0];

    const int ntiles = (E + 15) >> 4;
    const int wgid   = blockIdx.x * 8 + wave;
    const int nwv    = gridDim.x * 8;

    for (int t = wgid; t < ntiles; t += nwv) {
        const int e0 = t << 4;
        int e = e0 + m; if (e >= E) e = E - 1;     // clamp, keep EXEC full
        const int src = eidx[e];
        const int dst = eidx[E + e];
        const float* ps = emb + (size_t)src * EMB_D + g * 2;
        const float* pd = emb + (size_t)dst * EMB_D + g * 2;

        // ---- layer 1: [16x128] x [128x32], K in steps of 4 ----
        v8f c0 = {}, c1 = {};
        #pragma unroll
        for (int s = 0; s < 32; ++s) {
            v2f a  = *(const v2f*)(ps + 4 * s);
            v2f bb = *(const v2f*)(pd + 4 * s);
            v2f av; av[0] = a[0] * bb[0]; av[1] = a[1] * bb[1];   // feat frag
            v2f f0 = *(const v2f*)(&w1f[(s * 2 + 0) * 64 + lane * 2]);
            v2f f1 = *(const v2f*)(&w1f[(s * 2 + 1) * 64 + lane * 2]);
            c0 = __builtin_amdgcn_wmma_f32_16x16x4_f32(false, av, false, f0,
                                                       (short)0, c0, false, false);
            c1 = __builtin_amdgcn_wmma_f32_16x16x4_f32(false, av, false, f1,
                                                       (short)0, c1, false, false);
        }

        // bias + relu, transpose C/D layout -> row-major h1 in LDS (stride 34)
        #pragma unroll
        for (int v = 0; v < 8; ++v) {
            float x0 = c0[v] + b1lo; x0 = x0 > 0.f ? x0 : 0.f;
            float x1 = c1[v] + b1hi; x1 = x1 > 0.f ? x1 : 0.f;
            int row = v + 8 * g;
            h1[row * 34 + m]      = x0;
            h1[row * 34 + 16 + m] = x1;
        }
        // same-wave LDS ops are in-order (DScnt), no barrier needed

        // ---- layer 2: [16x32] x [32x32] ----
        v8f d0 = {}, d1 = {};
        #pragma unroll
        for (int s = 0; s < 8; ++s) {
            v2f a2 = *(const v2f*)(&h1[m * 34 + 4 * s + g * 2]);
            v2f f0 = *(const v2f*)(&w2f[(s * 2 + 0) * 64 + lane * 2]);
            v2f f1 = *(const v2f*)(&w2f[(s * 2 + 1) * 64 + lane * 2]);
            d0 = __builtin_amdgcn_wmma_f32_16x16x4_f32(false, a2, false, f0,
                                                       (short)0, d0, false, false);
            d1 = __builtin_amdgcn_wmma_f32_16x16x4_f32(false, a2, false, f1,
                                                       (short)0, d1, false, false);
        }

        // ---- layer 3: bias+relu then dot with W3, reduce across 16-lane halves
        #pragma unroll
        for (int v = 0; v < 8; ++v) {
            float x0 = d0[v] + b2lo; x0 = x0 > 0.f ? x0 : 0.f;
            float x1 = d1[v] + b2hi; x1 = x1 > 0.f ? x1 : 0.f;
            float tv = x0 * w3lo + x1 * w3hi;
            tv += __shfl_xor(tv, 1, 32);
            tv += __shfl_xor(tv, 2, 32);
            tv += __shfl_xor(tv, 4, 32);
            tv += __shfl_xor(tv, 8, 32);          // stays within each 16-lane half
            int eo = e0 + v + 8 * g;              // lanes 0..15 -> row v, 16..31 -> v+8
            if (m == 0 && eo < E) out[eo] = tv + b3s;
        }
    }
}

extern "C" void kernel_launch(void* const* d_in, const int* in_sizes, int n_in,
                              void* d_out, int out_size, void* d_ws, size_t ws_size,
                              hipStream_t stream) {
    // inputs: 0 node_id (identity, unused), 1 edge_label_index, 2 emb_table,
    //         3 W1, 4 b1, 5 W2, 6 b2, 7 W3, 8 b3
    const int*   eidx = (const int*)d_in[1];
    const float* emb  = (const float*)d_in[2];
    const float* W1   = (const float*)d_in[3];
    const float* b1   = (const float*)d_in[4];
    const float* W2   = (const float*)d_in[5];
    const float* b2   = (const float*)d_in[6];
    const float* W3   = (const float*)d_in[7];
    const float* b3   = (const float*)d_in[8];
    float* out = (float*)d_out;

    const int E = in_sizes[1] / 2;
    const int ntiles = (E + 15) >> 4;
    int blocks = (ntiles + 7) / 8;          // 8 wave-tiles per block pass
    if (blocks > 1024) blocks = 1024;       // persistent grid-stride over tiles
    if (blocks < 1)    blocks = 1;

    edge_mlp_wmma_f32<<<blocks, 256, 0, stream>>>(eidx, emb, W1, b1, W2, b2,
                                                  W3, b3, out, E);
}